// SimpleGNN_49941879717891
// MI455X (gfx1250) — compile-verified
//
#include <hip/hip_runtime.h>
#include <math.h>

typedef float v2f __attribute__((ext_vector_type(2)));
typedef float v8f __attribute__((ext_vector_type(8)));

#define HID 64

// ---------------- utility ----------------
__global__ void zero_kernel(float* __restrict__ p, int n) {
    int i = blockIdx.x * blockDim.x + threadIdx.x;
    if (i < n) p[i] = 0.0f;
}

__global__ void degree_kernel(const int* __restrict__ dst, float* __restrict__ deg, int E) {
    int e = blockIdx.x * blockDim.x + threadIdx.x;
    if (e < E) atomicAdd(&deg[dst[e]], 1.0f);
}

__global__ void rsqrt_kernel(float* __restrict__ dis, int N) {
    int i = blockIdx.x * blockDim.x + threadIdx.x;
    if (i < N) dis[i] = rsqrtf(dis[i] + 1.0f);
}

__global__ void norm_kernel(const int* __restrict__ src, const int* __restrict__ dst,
                            const float* __restrict__ dis, float* __restrict__ nrm, int E) {
    int e = blockIdx.x * blockDim.x + threadIdx.x;
    if (e < E) nrm[e] = dis[src[e]] * dis[dst[e]];
}

// ---------------- WMMA GEMM: layer 1, x[N,6] @ W1[6,64] -> A[N,64] (K padded 6->8)
__global__ void gemm1_wmma(const float* __restrict__ x, const float* __restrict__ W1,
                           float* __restrict__ A, int N) {
    int wid  = (blockIdx.x * blockDim.x + threadIdx.x) >> 5;  // global wave id
    int lane = threadIdx.x & 31;
    int row0 = wid * 16;
    if (row0 >= N) return;                 // wave-uniform: EXEC stays all-1 for WMMA
    int ln = lane & 15;                    // A: M index / B: N index
    int kb = (lane >> 4) << 1;             // half-wave K pair select: 0 or 2
    int r  = row0 + ln; if (r >= N) r = N - 1;

    v8f acc[4] = {};
    for (int k0 = 0; k0 < 8; k0 += 4) {
        int ka = k0 + kb;
        v2f a;
        a.x = (ka     < 6) ? x[(size_t)r * 6 + ka    ] : 0.0f;
        a.y = (ka + 1 < 6) ? x[(size_t)r * 6 + ka + 1] : 0.0f;
#pragma unroll
        for (int nt = 0; nt < 4; ++nt) {
            v2f b;
            b.x = (ka     < 6) ? W1[ ka      * HID + nt * 16 + ln] : 0.0f;
            b.y = (ka + 1 < 6) ? W1[(ka + 1) * HID + nt * 16 + ln] : 0.0f;
            acc[nt] = __builtin_amdgcn_wmma_f32_16x16x4_f32(
                false, a, false, b, (short)0, acc[nt], false, false);
        }
    }
    int rbase = row0 + ((lane >> 4) << 3);  // +8 rows for upper half-wave
#pragma unroll
    for (int nt = 0; nt < 4; ++nt)
#pragma unroll
        for (int i = 0; i < 8; ++i)
            if (rbase + i < N)
                A[(size_t)(rbase + i) * HID + nt * 16 + ln] = acc[nt][i];
}

// ---------------- WMMA GEMM: layer 2, H[N,64] @ W[64,64] -> H (in place; waves own disjoint rows)
__global__ void gemm2_wmma(float* __restrict__ H, const float* __restrict__ W, int N) {
    int wid  = (blockIdx.x * blockDim.x + threadIdx.x) >> 5;
    int lane = threadIdx.x & 31;
    int row0 = wid * 16;
    if (row0 >= N) return;
    int ln = lane & 15;
    int kb = (lane >> 4) << 1;
    int r  = row0 + ln; if (r >= N) r = N - 1;

    v8f acc[4] = {};
    for (int k0 = 0; k0 < HID; k0 += 4) {
        int ka = k0 + kb;
        v2f a;
        a.x = H[(size_t)r * HID + ka];
        a.y = H[(size_t)r * HID + ka + 1];
#pragma unroll
        for (int nt = 0; nt < 4; ++nt) {
            v2f b;
            b.x = W[ ka      * HID + nt * 16 + ln];
            b.y = W[(ka + 1) * HID + nt * 16 + ln];
            acc[nt] = __builtin_amdgcn_wmma_f32_16x16x4_f32(
                false, a, false, b, (short)0, acc[nt], false, false);
        }
    }
    int rbase = row0 + ((lane >> 4) << 3);
#pragma unroll
    for (int nt = 0; nt < 4; ++nt)
#pragma unroll
        for (int i = 0; i < 8; ++i)
            if (rbase + i < N)
                H[(size_t)(rbase + i) * HID + nt * 16 + ln] = acc[nt][i];
}

// ---------------- edge scatter: agg[dst] += h[src] * norm, 16 threads per edge (float4 each)
__global__ void scatter_kernel(const float* __restrict__ h, const int* __restrict__ src,
                               const int* __restrict__ dst, const float* __restrict__ nrm,
                               float* __restrict__ agg, int E) {
    unsigned tid = blockIdx.x * blockDim.x + threadIdx.x;
    unsigned e   = tid >> 4;
    unsigned c4  = tid & 15u;
    if (e >= (unsigned)E) return;
    int s = src[e], d = dst[e];
    float w = nrm[e];
    const float4 v = *reinterpret_cast<const float4*>(h + (size_t)s * HID + c4 * 4);
    float* o = agg + (size_t)d * HID + c4 * 4;
    atomicAdd(o + 0, v.x * w);
    atomicAdd(o + 1, v.y * w);
    atomicAdd(o + 2, v.z * w);
    atomicAdd(o + 3, v.w * w);
}

// ---------------- combine layer1: h = relu(agg + h*dis^2 + b1), in place on h
__global__ void combine1_kernel(float* __restrict__ h, const float* __restrict__ agg,
                                const float* __restrict__ dis, const float* __restrict__ b1, int N) {
    unsigned tid  = blockIdx.x * blockDim.x + threadIdx.x;
    unsigned node = tid >> 4, c4 = tid & 15u;
    if (node >= (unsigned)N) return;
    float d  = dis[node];
    float d2 = d * d;
    size_t off = (size_t)node * HID + c4 * 4;
    float4 hv = *reinterpret_cast<const float4*>(h + off);
    float4 av = *reinterpret_cast<const float4*>(agg + off);
    float4 r;
    r.x = fmaxf(av.x + hv.x * d2 + b1[c4 * 4 + 0], 0.0f);
    r.y = fmaxf(av.y + hv.y * d2 + b1[c4 * 4 + 1], 0.0f);
    r.z = fmaxf(av.z + hv.z * d2 + b1[c4 * 4 + 2], 0.0f);
    r.w = fmaxf(av.w + hv.w * d2 + b1[c4 * 4 + 3], 0.0f);
    *reinterpret_cast<float4*>(h + off) = r;
}

// ---------------- combine layer2 + per-channel column-sum reduction into g[64]
__global__ void combine2_reduce(const float* __restrict__ h, const float* __restrict__ agg,
                                const float* __restrict__ dis, const float* __restrict__ b2,
                                float* __restrict__ g, int N) {
    __shared__ float ls[HID];
    int t = threadIdx.x;
    if (t < HID) ls[t] = 0.0f;
    __syncthreads();
    unsigned tid  = blockIdx.x * blockDim.x + t;
    unsigned node = tid >> 4, c4 = tid & 15u;
    if (node < (unsigned)N) {
        float d  = dis[node];
        float d2 = d * d;
        size_t off = (size_t)node * HID + c4 * 4;
        float4 hv = *reinterpret_cast<const float4*>(h + off);
        float4 av = *reinterpret_cast<const float4*>(agg + off);
        atomicAdd(&ls[c4 * 4 + 0], av.x + hv.x * d2 + b2[c4 * 4 + 0]);
        atomicAdd(&ls[c4 * 4 + 1], av.y + hv.y * d2 + b2[c4 * 4 + 1]);
        atomicAdd(&ls[c4 * 4 + 2], av.z + hv.z * d2 + b2[c4 * 4 + 2]);
        atomicAdd(&ls[c4 * 4 + 3], av.w + hv.w * d2 + b2[c4 * 4 + 3]);
    }
    __syncthreads();
    if (t < HID) atomicAdd(&g[t], ls[t]);
}

// ---------------- readout: sigmoid(mean(out2) @ Wfc + bfc)
__global__ void final_kernel(const float* __restrict__ g, const float* __restrict__ Wfc,
                             const float* __restrict__ bfc, float* __restrict__ out, int N) {
    __shared__ float ls[HID];
    int t = threadIdx.x;  // 64 threads
    ls[t] = g[t] * Wfc[t];
    __syncthreads();
    for (int s = HID / 2; s > 0; s >>= 1) {
        if (t < s) ls[t] += ls[t + s];
        __syncthreads();
    }
    if (t == 0) {
        float z = ls[0] / (float)N + bfc[0];
        out[0] = 1.0f / (1.0f + expf(-z));
    }
}

extern "C" void kernel_launch(void* const* d_in, const int* in_sizes, int n_in,
                              void* d_out, int out_size, void* d_ws, size_t ws_size,
                              hipStream_t stream) {
    const float* x   = (const float*)d_in[0];
    const int*   ei  = (const int*)d_in[1];   // edge_index [2, E]
    const float* W1  = (const float*)d_in[2];
    const float* b1  = (const float*)d_in[3];
    const float* W2  = (const float*)d_in[4];
    const float* b2  = (const float*)d_in[5];
    const float* Wfc = (const float*)d_in[6];
    const float* bfc = (const float*)d_in[7];
    float* out = (float*)d_out;

    const int N = in_sizes[0] / 6;
    const int E = in_sizes[1] / 2;
    const int* src = ei;
    const int* dst = ei + E;

    // workspace layout (floats): dis[N] | A[N*64] | B[N*64] | g[64] | nrm[E]  (~57 MB)
    float* dis = (float*)d_ws;
    float* A   = dis + N;
    float* B   = A + (size_t)N * HID;
    float* g   = B + (size_t)N * HID;
    float* nrm = g + HID;

    const int T = 256;
    const int nf = N * HID;
    const int nwaves = (N + 15) / 16;
    const int gemm_blocks = (nwaves * 32 + T - 1) / T;
    const int edge_blocks = (int)(((size_t)E * 16 + T - 1) / T);
    const int node_blocks = (int)(((size_t)N * 16 + T - 1) / T);

    // degree + normalization
    zero_kernel  <<<(N + T - 1) / T, T, 0, stream>>>(dis, N);
    degree_kernel<<<(E + T - 1) / T, T, 0, stream>>>(dst, dis, E);
    rsqrt_kernel <<<(N + T - 1) / T, T, 0, stream>>>(dis, N);
    norm_kernel  <<<(E + T - 1) / T, T, 0, stream>>>(src, dst, dis, nrm, E);

    // layer 1
    gemm1_wmma    <<<gemm_blocks, T, 0, stream>>>(x, W1, A, N);
    zero_kernel   <<<(nf + HID + T - 1) / T, T, 0, stream>>>(B, nf + HID);  // agg + g
    scatter_kernel<<<edge_blocks, T, 0, stream>>>(A, src, dst, nrm, B, E);
    combine1_kernel<<<node_blocks, T, 0, stream>>>(A, B, dis, b1, N);

    // layer 2
    gemm2_wmma    <<<gemm_blocks, T, 0, stream>>>(A, W2, N);
    zero_kernel   <<<(nf + HID + T - 1) / T, T, 0, stream>>>(B, nf + HID);  // agg + g
    scatter_kernel<<<edge_blocks, T, 0, stream>>>(A, src, dst, nrm, B, E);
    combine2_reduce<<<node_blocks, T, 0, stream>>>(A, B, dis, b2, g, N);

    // readout
    final_kernel<<<1, HID, 0, stream>>>(g, Wfc, bfc, out, N);
}